// CustomGlobalAttention_52286931862219
// MI455X (gfx1250) — compile-verified
//
#include <hip/hip_runtime.h>
#include <hip/hip_bf16.h>

typedef __attribute__((ext_vector_type(2))) float v2f;
typedef __attribute__((ext_vector_type(4))) float v4f;
typedef __attribute__((ext_vector_type(8))) float v8f;

#define D_FEAT 256

// Monotonic float <-> uint32 key so atomicMax(u32) implements float max (incl. negatives).
__device__ __forceinline__ unsigned fkey(float f) {
  unsigned u = __float_as_uint(f);
  return u ^ ((u & 0x80000000u) ? 0xFFFFFFFFu : 0x80000000u);
}
__device__ __forceinline__ float funkey(unsigned k) {
  unsigned u = k ^ ((k & 0x80000000u) ? 0x80000000u : 0xFFFFFFFFu);
  return __uint_as_float(u);
}

__global__ void init_seg_kernel(unsigned* __restrict__ segmax,
                                float* __restrict__ segsum, int G) {
  int t = blockIdx.x * blockDim.x + threadIdx.x;
  if (t < G) { segmax[t] = 0u; segsum[t] = 0.0f; }
}

// Phase 1: gate logits = x @ W + b via V_WMMA_F32_16X16X4_F32.
// One wave handles 16 nodes; 64 K-steps of depth-4 over D=256, 4 accumulators.
__global__ void gate_wmma_kernel(const float* __restrict__ x,
                                 const long long* __restrict__ batch,
                                 const float* __restrict__ W,
                                 const float* __restrict__ bptr,
                                 float* __restrict__ gate,
                                 unsigned* __restrict__ segmax,
                                 int N) {
  const int wave = (int)((blockIdx.x * blockDim.x + threadIdx.x) >> 5);
  const int lane = threadIdx.x & 31;
  long m0 = (long)wave * 16;
  if (m0 >= N) return;                 // uniform per wave; EXEC stays full for WMMA

  const int khalf = lane >> 4;         // which K pair this lane holds
  long row = m0 + (lane & 15);
  if (row >= N) row = N - 1;           // safe clamp (N % 16 == 0 in practice)
  const float* xrow = x + row * (long)D_FEAT + 2 * khalf;
  const float bmask = ((lane & 15) == 0) ? 1.0f : 0.0f;   // only column N=0 of B is live

  v8f c0 = {}, c1 = {}, c2 = {}, c3 = {};
  auto step = [&](int kk, v8f c) -> v8f {
    v2f a = *(const v2f*)(xrow + kk);                  // A: K=kk+2*khalf, +1
    v2f w = *(const v2f*)(W + kk + 2 * khalf);         // B col 0: W[kk..kk+3]
    v2f b = { w.x * bmask, w.y * bmask };
    return __builtin_amdgcn_wmma_f32_16x16x4_f32(
        false, a, false, b, (short)0, c, false, false);
  };
  #pragma unroll
  for (int k0 = 0; k0 < D_FEAT; k0 += 16) {
    c0 = step(k0 +  0, c0);
    c1 = step(k0 +  4, c1);
    c2 = step(k0 +  8, c2);
    c3 = step(k0 + 12, c3);
  }
  v8f cs = (c0 + c1) + (c2 + c3);

  // Column N=0 lives in lane 0 (M=0..7) and lane 16 (M=8..15), one M per VGPR.
  if ((lane & 15) == 0) {
    const float bias = bptr[0];
    long base = m0 + (khalf ? 8 : 0);
    #pragma unroll
    for (int r = 0; r < 8; ++r) {
      long node = base + r;
      if (node < N) {
        float gl = cs[r] + bias;
        gate[node] = gl;
        int seg = (int)batch[node];
        atomicMax(&segmax[seg], fkey(gl));
      }
    }
  }
}

// Phase 2: e = exp(gate - segmax[seg]); gate <- e; segsum[seg] += e
__global__ void seg_exp_kernel(const long long* __restrict__ batch,
                               float* __restrict__ gate,
                               const unsigned* __restrict__ segmax,
                               float* __restrict__ segsum, int N) {
  int i = blockIdx.x * blockDim.x + threadIdx.x;
  if (i >= N) return;
  int seg = (int)batch[i];
  float mx = funkey(segmax[seg]);
  float e = expf(gate[i] - mx);
  gate[i] = e;
  atomicAdd(&segsum[seg], e);
}

__device__ __forceinline__ int lower_bound_ll(const long long* __restrict__ a,
                                              int n, long long v) {
  int lo = 0, hi = n;
  while (lo < hi) {
    int mid = (lo + hi) >> 1;
    if (a[mid] < v) lo = mid + 1; else hi = mid;
  }
  return lo;
}

// Phase 3: out[g] = sum_i (gate[i]/segsum[g]) * x[i]  over the sorted segment g.
// 256 threads = 4 node-rows x 64 float4 feature lanes; LDS reduce across rows.
__global__ void pool_kernel(const float* __restrict__ x,
                            const long long* __restrict__ batch,
                            const float* __restrict__ gate,
                            const float* __restrict__ segsum,
                            float* __restrict__ out, int N) {
  const int g  = blockIdx.x;
  const int t  = threadIdx.x;       // 256
  const int rg = t >> 6;            // row group 0..3
  const int d4 = (t & 63) << 2;     // feature offset (float4)

  int start = lower_bound_ll(batch, N, (long long)g);
  int end   = lower_bound_ll(batch, N, (long long)g + 1);
  float inv = 1.0f / (segsum[g] + 1e-16f);

  v4f acc = {};
  for (int i = start + rg; i < end; i += 4) {
    if (i + 4 < end) __builtin_prefetch(x + (long)(i + 4) * D_FEAT + d4, 0, 0);
    float w = gate[i] * inv;
    v4f v = *(const v4f*)(x + (long)i * D_FEAT + d4);
    acc += v * w;
  }

  __shared__ v4f sm[256];
  sm[t] = acc;
  __syncthreads();
  if (t < 64) {
    v4f s = (sm[t] + sm[t + 64]) + (sm[t + 128] + sm[t + 192]);
    *(v4f*)(out + (long)g * D_FEAT + (t << 2)) = s;
  }
}

extern "C" void kernel_launch(void* const* d_in, const int* in_sizes, int n_in,
                              void* d_out, int out_size, void* d_ws, size_t ws_size,
                              hipStream_t stream) {
  const float*     x      = (const float*)d_in[0];
  const long long* batch  = (const long long*)d_in[1];  // int64 per reference
  const float*     gate_W = (const float*)d_in[2];
  const float*     gate_b = (const float*)d_in[3];
  float*           out    = (float*)d_out;

  const int N = in_sizes[1];          // 100000 nodes
  const int G = out_size / D_FEAT;    // 512 segments

  // workspace layout: gate[N] | segmax_keys[G] | segsum[G]
  float* gate = (float*)d_ws;
  size_t off = ((size_t)N * sizeof(float) + 255) & ~(size_t)255;
  unsigned* segmax = (unsigned*)((char*)d_ws + off);
  float* segsum = (float*)(segmax + ((G + 63) & ~63));

  init_seg_kernel<<<(G + 255) / 256, 256, 0, stream>>>(segmax, segsum, G);

  int waves  = (N + 15) / 16;                       // 16 nodes per wave
  int blocks = (waves + 7) / 8;                     // 8 waves (256 thr) per block
  gate_wmma_kernel<<<blocks, 256, 0, stream>>>(x, batch, gate_W, gate_b,
                                               gate, segmax, N);

  seg_exp_kernel<<<(N + 255) / 256, 256, 0, stream>>>(batch, gate, segmax,
                                                      segsum, N);

  pool_kernel<<<G, 256, 0, stream>>>(x, batch, gate, segsum, out, N);
}